// CRFDecoder_9929964389119
// MI455X (gfx1250) — compile-verified
//
#include <hip/hip_runtime.h>

// ---------------------------------------------------------------------------
// CRF decoder for MI455X (gfx1250, wave32).
//  Kernel 1: feats = x @ W^T + b  via V_WMMA_F32_16X16X4_F32, double-buffered
//            LDS tiles fed by GLOBAL_LOAD_ASYNC_TO_LDS_B128 (ASYNCcnt path)
//            so HBM fetch of chunk c+1 overlaps WMMA compute of chunk c.
//            HBM-bound: 134 MB of x @ 23.3 TB/s ~= 5.8 us floor.
//  Kernel 2: Viterbi forward (wave-per-batch, shuffle max-plus) + LDS backtrace.
// ---------------------------------------------------------------------------

typedef float v2f __attribute__((ext_vector_type(2)));
typedef float v4f __attribute__((ext_vector_type(4)));
typedef float v8f __attribute__((ext_vector_type(8)));
typedef int   gv4i __attribute__((vector_size(16)));   // matches builtin param

#define HDIM 1024
#define TTAG 17
#define SEQ  512
#define BATCH 64
#define BK   64           // k-chunk staged in LDS
#define XPAD 68           // LDS row stride in floats (68 % 64 = 4 -> no bank conflicts)
#define NCHUNK (HDIM / BK)

#if defined(__AMDGCN__) && \
    __has_builtin(__builtin_amdgcn_global_load_async_to_lds_b128) && \
    __has_builtin(__builtin_amdgcn_s_wait_asynccnt)
#define USE_ASYNC 1
#else
#define USE_ASYNC 0
#endif

__device__ __forceinline__ void async_copy16(const float* g, float* l) {
#if USE_ASYNC
  __builtin_amdgcn_global_load_async_to_lds_b128(
      (__attribute__((address_space(1))) gv4i*)(void*)g,
      (__attribute__((address_space(3))) gv4i*)(void*)l,
      0, 0);
#else
  (void)g; (void)l;
#endif
}

// ---------------------------------------------------------------------------
// Kernel 1: GEMM. grid = (B*S/64) blocks x 128 threads (4 waves).
// Each wave owns a 16-row M-tile and both 16-wide N tiles (N=0..15, 16..31;
// only n<17 written back). K loop: 1024 in 16 chunks of 64; per chunk 16
// k4-steps x 2 WMMAs.
// ---------------------------------------------------------------------------
__global__ __launch_bounds__(128) void feats_gemm_wmma(
    const float* __restrict__ x,     // [B*S, H]
    const float* __restrict__ W,     // [T, H]
    const float* __restrict__ bias,  // [T]
    float* __restrict__ feats)       // [B*S, T]
{
  __shared__ float xs[2][64 * XPAD];   // double-buffered x tile (64 rows x 64 k)
  __shared__ float ws[2][32 * XPAD];   // double-buffered W tile (rows 17..31 zero)

  const int tid  = threadIdx.x;
  const int wave = tid >> 5;
  const int lane = tid & 31;
  const int row0 = blockIdx.x * 64;

  // Zero padded W rows 17..31 of BOTH buffers once (never overwritten later).
  for (int idx = tid; idx < (32 - TTAG) * XPAD; idx += 128) {
    ws[0][TTAG * XPAD + idx] = 0.0f;
    ws[1][TTAG * XPAD + idx] = 0.0f;
  }

  const int c4 = (tid & 15) * 4;   // float4 column within chunk (0..60)
  const int rb = tid >> 4;         // base row 0..7 (stride 8)

#if USE_ASYNC
  // Prologue: issue chunk 0 x-tile async loads into buffer 0 (8 per thread).
  #pragma unroll
  for (int rr = 0; rr < 64; rr += 8)
    async_copy16(x + (size_t)(row0 + rb + rr) * HDIM + c4,
                 &xs[0][(rb + rr) * XPAD + c4]);
#endif

  // W chunk staged via VGPRs (68 KB total, L2/WGP$-resident; tiny traffic).
  // 17 rows x 16 float4s = 272 elements; thread covers idx, idx+128, (idx+256 if tid<16).
  v4f wreg0, wreg1, wreg2;
  {
    const int i0 = tid, i1 = tid + 128;
    wreg0 = *(const v4f*)(W + (size_t)(i0 >> 4) * HDIM + (i0 & 15) * 4);
    wreg1 = *(const v4f*)(W + (size_t)(i1 >> 4) * HDIM + (i1 & 15) * 4);
    if (tid < 16) {
      const int i2 = tid + 256;
      wreg2 = *(const v4f*)(W + (size_t)(i2 >> 4) * HDIM + (i2 & 15) * 4);
    }
  }

  v8f acc0 = {};   // N tile 0 (cols 0..15)
  v8f acc1 = {};   // N tile 1 (cols 16..31; only col 16 valid)

  const int lr = lane & 15;          // A: row-in-tile; B: column n
  const int kb = (lane >> 4) << 1;   // 0 or 2 (K sub-offset per WMMA layout)

  for (int c = 0; c < NCHUNK; ++c) {
    const int cur = c & 1;
    const int nxt = cur ^ 1;

    __syncthreads();   // barrier A: all waves done reading buffers[nxt]

#if USE_ASYNC
    if (c + 1 < NCHUNK) {
      const int kc1 = (c + 1) * BK;
      #pragma unroll
      for (int rr = 0; rr < 64; rr += 8)
        async_copy16(x + (size_t)(row0 + rb + rr) * HDIM + kc1 + c4,
                     &xs[nxt][(rb + rr) * XPAD + c4]);
    }
#else
    {
      const int kc = c * BK;
      #pragma unroll
      for (int rr = 0; rr < 64; rr += 8)
        *(v4f*)(&xs[cur][(rb + rr) * XPAD + c4]) =
            *(const v4f*)(x + (size_t)(row0 + rb + rr) * HDIM + kc + c4);
    }
#endif

    // Store W chunk c into LDS; prefetch W chunk c+1 into VGPRs.
    {
      const int i0 = tid, i1 = tid + 128;
      *(v4f*)(&ws[cur][(i0 >> 4) * XPAD + (i0 & 15) * 4]) = wreg0;
      *(v4f*)(&ws[cur][(i1 >> 4) * XPAD + (i1 & 15) * 4]) = wreg1;
      if (tid < 16) {
        const int i2 = tid + 256;
        *(v4f*)(&ws[cur][(i2 >> 4) * XPAD + (i2 & 15) * 4]) = wreg2;
      }
      if (c + 1 < NCHUNK) {
        const int kc1 = (c + 1) * BK;
        wreg0 = *(const v4f*)(W + (size_t)(i0 >> 4) * HDIM + kc1 + (i0 & 15) * 4);
        wreg1 = *(const v4f*)(W + (size_t)(i1 >> 4) * HDIM + kc1 + (i1 & 15) * 4);
        if (tid < 16) {
          const int i2 = tid + 256;
          wreg2 = *(const v4f*)(W + (size_t)(i2 >> 4) * HDIM + kc1 + (i2 & 15) * 4);
        }
      }
    }

#if USE_ASYNC
    // ASYNCcnt is in-order per wave: <=8 outstanding guarantees chunk c landed
    // while chunk c+1's 8 transfers stay in flight (overlap with WMMAs below).
    if (c + 1 < NCHUNK) __builtin_amdgcn_s_wait_asynccnt(8);
    else                __builtin_amdgcn_s_wait_asynccnt(0);
#endif
    __syncthreads();   // barrier B: chunk-c x (async) + W (ds) visible to all

    const float* xa  = &xs[cur][(wave * 16 + lr) * XPAD + kb];
    const float* wb0 = &ws[cur][lr * XPAD + kb];
    const float* wb1 = &ws[cur][(16 + lr) * XPAD + kb];

    #pragma unroll
    for (int k4 = 0; k4 < BK; k4 += 4) {
      v2f a  = *(const v2f*)(xa  + k4);
      v2f b0 = *(const v2f*)(wb0 + k4);
      v2f b1 = *(const v2f*)(wb1 + k4);
      acc0 = __builtin_amdgcn_wmma_f32_16x16x4_f32(false, a, false, b0,
                                                   (short)0, acc0, false, false);
      acc1 = __builtin_amdgcn_wmma_f32_16x16x4_f32(false, a, false, b1,
                                                   (short)0, acc1, false, false);
    }
  }

  // Epilogue: C/D layout -> element r of lane: M = r + (lane>=16 ? 8:0), N = lane&15.
  const float bias0 = bias[lr];
  const bool  has1  = (lr == 0);   // N-tile-1: only col 16 valid
  const float bias1 = bias[16];
  const int   mbase = row0 + wave * 16 + ((lane >> 4) << 3);
  #pragma unroll
  for (int r = 0; r < 8; ++r) {
    const int row = mbase + r;
    feats[(size_t)row * TTAG + lr] = acc0[r] + bias0;
    if (has1) feats[(size_t)row * TTAG + 16] = acc1[r] + bias1;
  }
}

// ---------------------------------------------------------------------------
// Kernel 2: Viterbi. grid = 64 (one wave32 per batch), block = 32.
// Lane j owns tag j (j<17). Transition column trans[:,j] lives in registers.
// Backpointers in LDS (8.7 KB) so the backtrace chain runs at LDS latency.
// ---------------------------------------------------------------------------
__global__ __launch_bounds__(32) void viterbi_decode(
    const float* __restrict__ feats,       // [B, S, T]
    const float* __restrict__ trans,       // [T, T]
    const float* __restrict__ start_trans, // [T]
    const float* __restrict__ end_trans,   // [T]
    const int*   __restrict__ nwords,      // [B]
    float* __restrict__ tags_out)          // [B, S]
{
  __shared__ unsigned char bp[(SEQ - 1) * TTAG];   // bp for steps t=1..511

  const int b    = blockIdx.x;
  const int lane = threadIdx.x;
  const int j    = lane;
  const int jj   = (j < TTAG) ? j : (TTAG - 1);    // clamp for safe loads
  const size_t fbase = (size_t)b * SEQ * TTAG;

  // Transition column for my tag: tcol[i] = trans[i][j].
  float tcol[TTAG];
  #pragma unroll
  for (int i = 0; i < TTAG; ++i) tcol[i] = trans[i * TTAG + jj];

  const int n = nwords[b];

  float s = start_trans[jj] + feats[fbase + jj];   // init at position 0
  float f_next = feats[fbase + TTAG + jj];         // prefetch position 1

  for (int t = 1; t < SEQ; ++t) {
    const float f = f_next;
    if (t + 1 < SEQ) f_next = feats[fbase + (size_t)(t + 1) * TTAG + jj];

    // best_j = max_i(score[i] + trans[i][j]); first-index argmax.
    float best = -3.402823466e+38f;
    int   bi   = 0;
    #pragma unroll
    for (int i = 0; i < TTAG; ++i) {
      const float c = __shfl(s, i, 32) + tcol[i];
      if (c > best) { best = c; bi = i; }
    }

    const bool m = (t < n);
    const float ns  = m ? (best + f) : s;
    const int   nbp = m ? bi : j;
    if (j < TTAG) bp[(t - 1) * TTAG + j] = (unsigned char)nbp;
    s = ns;
  }

  // last_tag = argmax_j(score[j] + end_trans[j]), first-index tie-break.
  const float fin = s + end_trans[jj];
  int last = 0;
  float bv = -3.402823466e+38f;
  #pragma unroll
  for (int i = 0; i < TTAG; ++i) {
    const float v = __shfl(fin, i, 32);
    if (v > bv) { bv = v; last = i; }
  }

  __syncthreads();  // make bp[] visible to lane 0

  if (lane == 0) {
    float* out = tags_out + (size_t)b * SEQ;
    int tag = last;
    out[SEQ - 1] = (SEQ - 1 < n) ? (float)tag : 0.0f;
    for (int t = SEQ - 2; t >= 0; --t) {
      tag = bp[t * TTAG + tag];                 // tags[t] = bps[t][tags[t+1]]
      out[t] = (t < n) ? (float)tag : 0.0f;
    }
  }
}

// ---------------------------------------------------------------------------
extern "C" void kernel_launch(void* const* d_in, const int* in_sizes, int n_in,
                              void* d_out, int out_size, void* d_ws, size_t ws_size,
                              hipStream_t stream) {
  const float* x     = (const float*)d_in[0];  // [64,512,1024]
  const float* W     = (const float*)d_in[1];  // [17,1024]
  const float* bias  = (const float*)d_in[2];  // [17]
  const float* trans = (const float*)d_in[3];  // [17,17]
  const float* st    = (const float*)d_in[4];  // [17]
  const float* et    = (const float*)d_in[5];  // [17]
  const int*   nw    = (const int*)d_in[6];    // [64]

  float* tags  = (float*)d_out;                         // [64,512]
  float* feats = (float*)d_out + (size_t)BATCH * SEQ;   // [64,512,17]

  feats_gemm_wmma<<<(BATCH * SEQ) / 64, 128, 0, stream>>>(x, W, bias, feats);
  viterbi_decode<<<BATCH, 32, 0, stream>>>(feats, trans, st, et, nw, tags);
}